// Model_39436389711917
// MI455X (gfx1250) — compile-verified
//
#include <hip/hip_runtime.h>
#include <hip/hip_bf16.h>
#include <stdint.h>

#define NNODES   400000
#define NEDGES   3200000
#define KCHEB    5
#define HDIM     100
#define F_IN     2
#define NLAYERS  5
#define BN_EPS_F 1e-5f
#define OSTR     112      // padded column stride of the GEMM accumulator
#define NCT      7        // 7 column tiles of 16 cover 112 >= 100

typedef __attribute__((ext_vector_type(2))) float v2f;
typedef __attribute__((ext_vector_type(8))) float v8f;

// ---------------------------------------------------------------- utilities

__global__ void zero4_kernel(float4* __restrict__ p, long long n4) {
    long long i = (long long)blockIdx.x * blockDim.x + threadIdx.x;
    long long stride = (long long)gridDim.x * blockDim.x;
    float4 z = make_float4(0.f, 0.f, 0.f, 0.f);
    for (; i < n4; i += stride) p[i] = z;
}

__global__ void degree_kernel(float* __restrict__ deg,
                              const long long* __restrict__ src, int e_count) {
    int e = blockIdx.x * blockDim.x + threadIdx.x;
    if (e < e_count) atomicAdd(&deg[src[e]], 1.0f);
}

__global__ void dinv_kernel(float* __restrict__ dinv,
                            const float* __restrict__ deg, int n) {
    int i = blockIdx.x * blockDim.x + threadIdx.x;
    if (i < n) {
        float d = deg[i];
        dinv[i] = (d > 0.0f) ? rsqrtf(fmaxf(d, 1.0f)) : 0.0f;
    }
}

__global__ void edge_norm_kernel(float* __restrict__ nrm,
                                 const float* __restrict__ dinv,
                                 const long long* __restrict__ src,
                                 const long long* __restrict__ dst, int e_count) {
    int e = blockIdx.x * blockDim.x + threadIdx.x;
    if (e < e_count) nrm[e] = -dinv[src[e]] * dinv[dst[e]];
}

// ------------------------------------------------------ sparse propagation
// out[dst] += norm_e * h[src]; one wave32 per edge, lane-strided features.
// Edge index is forced wave-uniform (readfirstlane) so src/dst/norm become
// scalar (SMEM) loads and the gather addresses use the saddr form. The 160MB
// feature matrix fits MI455X's 192MB L2, so gathers are L2 hits and the
// scatter-adds resolve in L2 atomic units.
__global__ __launch_bounds__(256)
void prop_kernel(float* __restrict__ outp, const float* __restrict__ h,
                 const long long* __restrict__ src, const long long* __restrict__ dst,
                 const float* __restrict__ nrm, int e_count, int F) {
    int e = __builtin_amdgcn_readfirstlane(blockIdx.x * 8 + (threadIdx.x >> 5));
    if (e >= e_count) return;
    int lane = threadIdx.x & 31;
    long long s = src[e];
    long long d = dst[e];
    float w = nrm[e];
    const float* __restrict__ hrow = h + s * (long long)F;
    float* __restrict__ orow = outp + d * (long long)F;
    for (int f = lane; f < F; f += 32)
        atomicAdd(&orow[f], w * hrow[f]);
}

// Chebyshev recurrence combine: B = 2*B - A   (float4-vectorized)
__global__ void cheb_combine_kernel(float4* __restrict__ b,
                                    const float4* __restrict__ a, long long n4) {
    long long i = (long long)blockIdx.x * blockDim.x + threadIdx.x;
    long long stride = (long long)gridDim.x * blockDim.x;
    for (; i < n4; i += stride) {
        float4 bv = b[i], av = a[i];
        bv.x = 2.0f * bv.x - av.x;
        bv.y = 2.0f * bv.y - av.y;
        bv.z = 2.0f * bv.z - av.z;
        bv.w = 2.0f * bv.w - av.w;
        b[i] = bv;
    }
}

// ------------------------------------------------------------ WMMA GEMM
// C[N x OSTR] (+)= A[N x FIN] @ Wk[FIN x 100]  (+ bias on first call)
// fp32 tensor path: V_WMMA_F32_16X16X4_F32. One wave computes a 16x112
// row-strip (7 tiles, 7 accumulators), reusing each A fragment across 7
// WMMAs. W is staged in LDS pair-interleaved so each B fragment is one
// ds_load_b64; C is stride-112 padded so all C traffic is unguarded.
// Fragment layouts per ISA 7.12.2:
//   A 16x4 f32: lanes 0-15 hold K={0,1}, lanes 16-31 K={2,3}, M = lane&15
//   B 4x16 mirrors A; C/D: vgpr v holds M = v + 8*(lane>=16), N = lane&15
template<int FIN>
__global__ __launch_bounds__(256)
void gemm_wmma_kernel(float* __restrict__ C, const float* __restrict__ A,
                      const float* __restrict__ Wk, const float* __restrict__ bias,
                      int accum) {
    constexpr int PADK = ((FIN + 3) / 4) * 4;
    __shared__ float sW[(PADK / 2) * (2 * OSTR)];   // pair-interleaved panel

    const int tid   = threadIdx.x;
    const int lane  = tid & 31;
    const int wid   = tid >> 5;
    const int rb    = blockIdx.x * 128 + wid * 16;  // 400000 % 128 == 0
    const int c     = lane & 15;
    const int khalf = (lane >> 4) << 1;             // 0 | 2
    const int hi8   = (lane >> 4) << 3;

    // stage W: sW[(k>>1)*(2*OSTR) + col*2 + (k&1)] = W[k][col] (zero-padded)
    for (int i = tid; i < PADK * OSTR; i += 256) {
        int k = i / OSTR, col = i % OSTR;
        int kc = (k < FIN) ? k : 0;
        int cc = (col < HDIM) ? col : 0;
        float v = Wk[kc * HDIM + cc];
        v = (k < FIN && col < HDIM) ? v : 0.0f;
        sW[(k >> 1) * (2 * OSTR) + col * 2 + (k & 1)] = v;
    }
    __syncthreads();

    const float* __restrict__ ap = A + (size_t)(rb + c) * FIN;
    float* __restrict__ crow = C + (size_t)(rb + hi8) * OSTR + c;

    v8f acc[NCT];
    if (accum) {
#pragma unroll
        for (int t = 0; t < NCT; ++t)
#pragma unroll
            for (int v = 0; v < 8; ++v)
                acc[t][v] = crow[(size_t)v * OSTR + t * 16];
    } else {
#pragma unroll
        for (int t = 0; t < NCT; ++t) {
            int col = t * 16 + c;
            int cc = (col < HDIM) ? col : 0;      // clamp addr, select value:
            float bv = bias[cc];                  // no exec-mask branches
            bv = (col < HDIM) ? bv : 0.0f;
#pragma unroll
            for (int v = 0; v < 8; ++v) acc[t][v] = bv;
        }
    }

    const float* lbase = &sW[c * 2];

    if constexpr ((FIN & 3) == 0) {
        // all k-guards statically removed; A frag = one global_load_b64
        v2f a = *(const v2f*)(ap + khalf);
        for (int k0 = 0; k0 < FIN; k0 += 4) {
            v2f a_nxt = a;
            if (k0 + 4 < FIN)                      // wave-uniform branch
                a_nxt = *(const v2f*)(ap + k0 + 4 + khalf);
            const float* lp = lbase + ((k0 + khalf) >> 1) * (2 * OSTR);
#pragma unroll
            for (int t = 0; t < NCT; ++t) {
                v2f b = *(const v2f*)(lp + t * 32);
                acc[t] = __builtin_amdgcn_wmma_f32_16x16x4_f32(
                             false, a, false, b, (short)0, acc[t], false, false);
            }
            a = a_nxt;
        }
    } else {
        // FIN == 2: single k-step; upper lane-half (K=2,3) contributes zero.
        bool ok = (khalf < FIN);
        float a0 = ap[0];
        float a1 = ap[FIN > 1 ? 1 : 0];
        v2f a;
        a.x = ok ? a0 : 0.0f;
        a.y = ok ? a1 : 0.0f;
        const float* lp = lbase + (khalf >> 1) * (2 * OSTR);  // zero-padded rows
#pragma unroll
        for (int t = 0; t < NCT; ++t) {
            v2f b = *(const v2f*)(lp + t * 32);
            acc[t] = __builtin_amdgcn_wmma_f32_16x16x4_f32(
                         false, a, false, b, (short)0, acc[t], false, false);
        }
    }

#pragma unroll
    for (int t = 0; t < NCT; ++t)
#pragma unroll
        for (int v = 0; v < 8; ++v)
            crow[(size_t)v * OSTR + t * 16] = acc[t][v];
}

// --------------------------------------------------------- BN stats + apply
// Pass 1: per-column sum / sumsq of relu(X) (X has stride OSTR).
__global__ __launch_bounds__(128)
void relu_stats_kernel(const float* __restrict__ X, float* __restrict__ stats,
                       int n_rows, int rows_per_block) {
    int col = threadIdx.x;
    if (col >= HDIM) return;
    long long r0 = (long long)blockIdx.x * rows_per_block;
    long long r1 = r0 + rows_per_block;
    if (r1 > n_rows) r1 = n_rows;
    float s = 0.0f, ss = 0.0f;
    for (long long r = r0; r < r1; ++r) {
        float v = fmaxf(X[r * OSTR + col], 0.0f);
        s += v;
        ss += v * v;
    }
    atomicAdd(&stats[col], s);
    atomicAdd(&stats[HDIM + col], ss);
}

// Pass 2: out[r*HDIM+c] = gamma*(relu(X)-mean)*rsqrt(var+eps)+beta
__global__ __launch_bounds__(128)
void bn_apply_kernel(float* __restrict__ outp, const float* __restrict__ X,
                     const float* __restrict__ stats,
                     const float* __restrict__ gamma,
                     const float* __restrict__ beta,
                     int n_rows, int rows_per_block) {
    int col = threadIdx.x;
    if (col >= HDIM) return;
    const float inv_n = 1.0f / (float)NNODES;
    float m  = stats[col] * inv_n;
    float var = stats[HDIM + col] * inv_n - m * m;
    float sc = gamma[col] * rsqrtf(var + BN_EPS_F);
    float sh = beta[col] - m * sc;
    long long r0 = (long long)blockIdx.x * rows_per_block;
    long long r1 = r0 + rows_per_block;
    if (r1 > n_rows) r1 = n_rows;
    for (long long r = r0; r < r1; ++r) {
        float v = fmaxf(X[r * OSTR + col], 0.0f);
        outp[r * HDIM + col] = sc * v + sh;
    }
}

// ------------------------------------------------------------------ driver

extern "C" void kernel_launch(void* const* d_in, const int* in_sizes, int n_in,
                              void* d_out, int out_size, void* d_ws, size_t ws_size,
                              hipStream_t stream) {
    const float*     x     = (const float*)d_in[0];
    const long long* ei    = (const long long*)d_in[1];   // int64 in reference
    const float*     W0    = (const float*)d_in[2];       // [K, F_IN, H]
    const float*     b0    = (const float*)d_in[3];       // [H]
    const float*     Wr    = (const float*)d_in[4];       // [L-1, K, H, H]
    const float*     br    = (const float*)d_in[5];       // [L-1, H]
    const float*     gamma = (const float*)d_in[6];       // [L, H]
    const float*     beta  = (const float*)d_in[7];       // [L, H]
    float*           out   = (float*)d_out;

    const long long* src = ei;
    const long long* dst = ei + NEDGES;

    // workspace carve (floats); every buffer offset is 16B aligned
    float* p = (float*)d_ws;
    float* deg   = p;                 p += NNODES;
    float* dinv  = p;                 p += NNODES;
    float* enorm = p;                 p += NEDGES;
    float* stats = p;                 p += 2 * HDIM;
    const size_t NH = (size_t)NNODES * HDIM;     // T-buffer size (stride 100)
    const size_t NO = (size_t)NNODES * OSTR;     // accumulator (stride 112)
    float* bufA = p;                  p += NH;   // T1 / T4
    float* bufB = p;                  p += NH;   // T2
    float* bufC = p;                  p += NH;   // T3
    float* bufO = p;                  p += NO;   // GEMM accumulator (padded)
    float* bufH = p;                  p += NH;   // layer output (h)

    const int TB = 256;
    int gemm_blocks = (NNODES + 127) / 128;      // 3125, full strip per wave
    int prop_blocks = (NEDGES + 7) / 8;          // 8 edges per 256-thread block
    int row_chunk = 512;
    int row_blocks = (NNODES + row_chunk - 1) / row_chunk;

    // --- graph normalization (once) ---
    zero4_kernel<<<1024, TB, 0, stream>>>((float4*)deg, NNODES / 4);
    degree_kernel<<<(NEDGES + TB - 1) / TB, TB, 0, stream>>>(deg, src, NEDGES);
    dinv_kernel<<<(NNODES + TB - 1) / TB, TB, 0, stream>>>(dinv, deg, NNODES);
    edge_norm_kernel<<<(NEDGES + TB - 1) / TB, TB, 0, stream>>>(enorm, dinv, src, dst, NEDGES);

    const float* hcur = x;

    for (int layer = 0; layer < NLAYERS; ++layer) {
        const int Fin = (layer == 0) ? F_IN : HDIM;
        const float* Wbase;
        const float* bias;
        size_t wstride;
        if (layer == 0) {
            Wbase = W0; bias = b0; wstride = (size_t)F_IN * HDIM;
        } else {
            Wbase = Wr + (size_t)(layer - 1) * KCHEB * HDIM * HDIM;
            bias  = br + (size_t)(layer - 1) * HDIM;
            wstride = (size_t)HDIM * HDIM;
        }
        long long nF4 = (long long)NNODES * Fin / 4;

        auto gemm = [&](const float* Amat, const float* Wk, const float* bs, int accum) {
            if (layer == 0)
                gemm_wmma_kernel<F_IN><<<gemm_blocks, TB, 0, stream>>>(bufO, Amat, Wk, bs, accum);
            else
                gemm_wmma_kernel<HDIM><<<gemm_blocks, TB, 0, stream>>>(bufO, Amat, Wk, bs, accum);
        };

        // k=0: O = T0 @ W[0] + bias          (T0 = hcur)
        gemm(hcur, Wbase, bias, 0);

        // k=1: T1 = prop(T0); O += T1 @ W[1]
        zero4_kernel<<<2048, TB, 0, stream>>>((float4*)bufA, nF4);
        prop_kernel<<<prop_blocks, TB, 0, stream>>>(bufA, hcur, src, dst, enorm, NEDGES, Fin);
        gemm(bufA, Wbase + 1 * wstride, bias, 1);

        // k=2: T2 = 2*prop(T1) - T0 -> bufB
        zero4_kernel<<<2048, TB, 0, stream>>>((float4*)bufB, nF4);
        prop_kernel<<<prop_blocks, TB, 0, stream>>>(bufB, bufA, src, dst, enorm, NEDGES, Fin);
        cheb_combine_kernel<<<2048, TB, 0, stream>>>((float4*)bufB, (const float4*)hcur, nF4);
        gemm(bufB, Wbase + 2 * wstride, bias, 1);

        // k=3: T3 = 2*prop(T2) - T1 -> bufC
        zero4_kernel<<<2048, TB, 0, stream>>>((float4*)bufC, nF4);
        prop_kernel<<<prop_blocks, TB, 0, stream>>>(bufC, bufB, src, dst, enorm, NEDGES, Fin);
        cheb_combine_kernel<<<2048, TB, 0, stream>>>((float4*)bufC, (const float4*)bufA, nF4);
        gemm(bufC, Wbase + 3 * wstride, bias, 1);

        // k=4: T4 = 2*prop(T3) - T2 -> bufA (reused)
        zero4_kernel<<<2048, TB, 0, stream>>>((float4*)bufA, nF4);
        prop_kernel<<<prop_blocks, TB, 0, stream>>>(bufA, bufC, src, dst, enorm, NEDGES, Fin);
        cheb_combine_kernel<<<2048, TB, 0, stream>>>((float4*)bufA, (const float4*)bufB, nF4);
        gemm(bufA, Wbase + 4 * wstride, bias, 1);

        // ReLU + BatchNorm
        zero4_kernel<<<1, 64, 0, stream>>>((float4*)stats, (2 * HDIM) / 4);
        relu_stats_kernel<<<row_blocks, 128, 0, stream>>>(bufO, stats, NNODES, row_chunk);
        float* tgt = (layer == NLAYERS - 1) ? out : bufH;
        bn_apply_kernel<<<row_blocks, 128, 0, stream>>>(tgt, bufO, stats,
                                                        gamma + (size_t)layer * HDIM,
                                                        beta + (size_t)layer * HDIM,
                                                        NNODES, row_chunk);
        hcur = bufH;
    }
}